// GeometryRefiner_49117245997571
// MI455X (gfx1250) — compile-verified
//
#include <hip/hip_runtime.h>
#include <math.h>

#define S_RES   256
#define A_AT    5
#define NATOM   (S_RES * A_AT)        // 1280 atoms per batch
#define NB      2
#define NCOORD  (NB * NATOM * 3)      // 7680 floats
#define N_TILES (NATOM / 16)          // 80 clash row/col tiles
#define NR_TILES (S_RES / 16)         // 16 restraint tiles

#define CLASH_T2  4.0f                // CLASH_THRESH^2
#define TGT_BOND  1.5f
#define TGT_ANG   1.9111355309337906f // 109.5 deg in rad

#define NBOND_PB  1275                // (S-1)*(A-1) + (S-1)
#define NBONDS    (NB * NBOND_PB)     // 2550
#define NANG_PB   (S_RES * 3)         // 768
#define NANGS     (NB * NANG_PB)      // 1536

typedef __attribute__((ext_vector_type(2))) float v2f;
typedef __attribute__((ext_vector_type(8))) float v8f;

// ---------------------------------------------------------------------------
// init: copy coords into workspace x, zero g/m/v
// ---------------------------------------------------------------------------
__global__ void gr_init(const float* __restrict__ coords,
                        float* __restrict__ xw, float* __restrict__ gw,
                        float* __restrict__ mw, float* __restrict__ vw) {
    int i = blockIdx.x * blockDim.x + threadIdx.x;
    if (i < NCOORD) { xw[i] = coords[i]; gw[i] = 0.0f; mw[i] = 0.0f; vw[i] = 0.0f; }
}

// ---------------------------------------------------------------------------
// clash gradient: O(N^2) pairwise, dot products via v_wmma_f32_16x16x4_f32.
// One wave owns one 16-row tile; loops over all 80 column tiles.
// Active pair: pm && 1e-12 < d2 < 4;  w = 1/d = rsqrt(d2)  (no sqrt needed).
// grad_i = cscale * (sum_j w*x_j - W*x_i), cscale = 2*W_CLASH/(B*n_valid).
// ---------------------------------------------------------------------------
__global__ __launch_bounds__(128) void gr_clash_grad(const float* __restrict__ x,
                                                     float* __restrict__ g,
                                                     float cscale) {
    __shared__ float4 sA[NATOM];                 // (x,y,z,|x|^2), 20KB LDS
    const int b     = blockIdx.y;
    const int wave  = threadIdx.x >> 5;
    const int lane  = threadIdx.x & 31;
    const int l15   = lane & 15;
    const int hi    = lane >> 4;                 // lane group 0/1
    const int tileI = blockIdx.x * 4 + wave;     // 0..79

    const float* xb = x + b * NATOM * 3;
    for (int n = threadIdx.x; n < NATOM; n += 128) {
        float cx = xb[3*n], cy = xb[3*n+1], cz = xb[3*n+2];
        sA[n] = make_float4(cx, cy, cz, cx*cx + cy*cy + cz*cz);
    }
    __syncthreads();

    // A fragment: 16x4 f32 layout -> lane l holds row (l&15), K={0,1} lo / {2,3} hi
    float4 am = sA[tileI*16 + l15];
    v2f afrag;
    afrag.x = hi ? am.z : am.x;
    afrag.y = hi ? 0.0f : am.y;

    // C-row metadata: VGPR r holds row r (lanes 0-15) / r+8 (lanes 16-31)
    float rowSq[8]; int rowRes[8];
    #pragma unroll
    for (int r = 0; r < 8; ++r) {
        int ia = tileI*16 + r + 8*hi;
        rowSq[r]  = sA[ia].w;
        rowRes[r] = ia / A_AT;
    }

    float accW[8], accX[8], accY[8], accZ[8];
    #pragma unroll
    for (int r = 0; r < 8; ++r) { accW[r]=0.f; accX[r]=0.f; accY[r]=0.f; accZ[r]=0.f; }

    for (int tj = 0; tj < N_TILES; ++tj) {
        int ja = tj*16 + l15;
        float4 aj = sA[ja];
        v2f bfrag;
        bfrag.x = hi ? aj.z : aj.x;
        bfrag.y = hi ? 0.0f : aj.y;
        v8f c = {};
        c = __builtin_amdgcn_wmma_f32_16x16x4_f32(false, afrag, false, bfrag,
                                                  (short)0, c, false, false);
        float sqj  = aj.w;
        int   resj = ja / A_AT;
        #pragma unroll
        for (int r = 0; r < 8; ++r) {
            float d2 = rowSq[r] + sqj - 2.0f * c[r];
            int   dr = rowRes[r] - resj;
            bool  pm = (dr > 1) || (dr < -1);
            bool  act = pm && (d2 > 1e-12f) && (d2 < CLASH_T2);
            float w  = act ? __builtin_amdgcn_rsqf(d2) : 0.0f;   // 1/d
            accW[r] += w;
            accX[r] += w * aj.x;
            accY[r] += w * aj.y;
            accZ[r] += w * aj.z;
        }
    }

    // reduce across the 16 lanes holding each row (xor<16 stays in-half)
    #pragma unroll
    for (int r = 0; r < 8; ++r) {
        float W = accW[r], X = accX[r], Y = accY[r], Z = accZ[r];
        #pragma unroll
        for (int off = 1; off < 16; off <<= 1) {
            W += __shfl_xor(W, off, 32);
            X += __shfl_xor(X, off, 32);
            Y += __shfl_xor(Y, off, 32);
            Z += __shfl_xor(Z, off, 32);
        }
        if (l15 == 0) {
            int ia = tileI*16 + r + 8*hi;
            float4 xi = sA[ia];
            float* gp = g + (b*NATOM + ia)*3;
            atomicAdd(gp+0, cscale*(X - W*xi.x));
            atomicAdd(gp+1, cscale*(Y - W*xi.y));
            atomicAdd(gp+2, cscale*(Z - W*xi.z));
        }
    }
}

// ---------------------------------------------------------------------------
// restraint gradient over rep atoms (atom 0 of each residue), same WMMA scheme.
// w = 2*(d - R')/d = 2 - 2*R'*rsqrt(d2)  (no sqrt needed).
// grad_i = rscale * (W*x_i - sum_j w*x_j).
// ---------------------------------------------------------------------------
__global__ __launch_bounds__(128) void gr_rest_grad(const float* __restrict__ x,
                                                    const float* __restrict__ restr,
                                                    float* __restrict__ g,
                                                    float rscale) {
    __shared__ float4 sR[S_RES];                 // 4KB LDS
    const int b     = blockIdx.y;
    const int wave  = threadIdx.x >> 5;
    const int lane  = threadIdx.x & 31;
    const int l15   = lane & 15;
    const int hi    = lane >> 4;
    const int tileI = blockIdx.x * 4 + wave;     // 0..15

    const float* xb = x + b * NATOM * 3;
    for (int n = threadIdx.x; n < S_RES; n += 128) {
        const float* p = xb + n * A_AT * 3;      // atom 0 of residue n
        float cx = p[0], cy = p[1], cz = p[2];
        sR[n] = make_float4(cx, cy, cz, cx*cx + cy*cy + cz*cz);
    }
    __syncthreads();

    float4 am = sR[tileI*16 + l15];
    v2f afrag;
    afrag.x = hi ? am.z : am.x;
    afrag.y = hi ? 0.0f : am.y;

    float rowSq[8]; int rowI[8];
    #pragma unroll
    for (int r = 0; r < 8; ++r) {
        rowI[r]  = tileI*16 + r + 8*hi;
        rowSq[r] = sR[rowI[r]].w;
    }

    const float* Rb = restr + b * S_RES * S_RES;

    float accW[8], accX[8], accY[8], accZ[8];
    #pragma unroll
    for (int r = 0; r < 8; ++r) { accW[r]=0.f; accX[r]=0.f; accY[r]=0.f; accZ[r]=0.f; }

    for (int tj = 0; tj < NR_TILES; ++tj) {
        int jr = tj*16 + l15;
        float4 aj = sR[jr];
        v2f bfrag;
        bfrag.x = hi ? aj.z : aj.x;
        bfrag.y = hi ? 0.0f : aj.y;
        v8f c = {};
        c = __builtin_amdgcn_wmma_f32_16x16x4_f32(false, afrag, false, bfrag,
                                                  (short)0, c, false, false);
        #pragma unroll
        for (int r = 0; r < 8; ++r) {
            int  i  = rowI[r];
            int  lo = (i < jr) ? i : jr;
            int  ho = (i < jr) ? jr : i;
            float Rv = Rb[lo * S_RES + ho];      // R' (upper-tri entry)
            float d2 = rowSq[r] + aj.w - 2.0f * c[r];
            bool  act = (i != jr) && (d2 > 1e-12f);
            float w  = act ? 2.0f - 2.0f * Rv * __builtin_amdgcn_rsqf(d2) : 0.0f;
            accW[r] += w;
            accX[r] += w * aj.x;
            accY[r] += w * aj.y;
            accZ[r] += w * aj.z;
        }
    }

    #pragma unroll
    for (int r = 0; r < 8; ++r) {
        float W = accW[r], X = accX[r], Y = accY[r], Z = accZ[r];
        #pragma unroll
        for (int off = 1; off < 16; off <<= 1) {
            W += __shfl_xor(W, off, 32);
            X += __shfl_xor(X, off, 32);
            Y += __shfl_xor(Y, off, 32);
            Z += __shfl_xor(Z, off, 32);
        }
        if (l15 == 0) {
            int i = rowI[r];
            float4 xi = sR[i];
            float* gp = g + (b*NATOM + i*A_AT)*3;    // rep atom = atom 0
            atomicAdd(gp+0, rscale*(W*xi.x - X));
            atomicAdd(gp+1, rscale*(W*xi.y - Y));
            atomicAdd(gp+2, rscale*(W*xi.z - Z));
        }
    }
}

// ---------------------------------------------------------------------------
// bond + angle gradients (tiny, atomic accumulation)
// ---------------------------------------------------------------------------
__global__ void gr_local_grad(const float* __restrict__ x, float* __restrict__ g,
                              float bscale, float ascale) {
    int tid = blockIdx.x * blockDim.x + threadIdx.x;
    if (tid < NBONDS) {
        int b = tid / NBOND_PB;
        int r = tid % NBOND_PB;
        int n1, n2;
        if (r < (S_RES-1)*(A_AT-1)) {            // intra bonds, residues 0..S-2
            int s = r >> 2; int a = r & 3;
            n1 = s*A_AT + a; n2 = n1 + 1;
        } else {                                  // inter bonds
            int s = r - (S_RES-1)*(A_AT-1);
            n1 = s*A_AT + (A_AT-1); n2 = (s+1)*A_AT;
        }
        const float* p1 = x + (b*NATOM + n1)*3;
        const float* p2 = x + (b*NATOM + n2)*3;
        float dx = p2[0]-p1[0], dy = p2[1]-p1[1], dz = p2[2]-p1[2];
        float l2 = dx*dx + dy*dy + dz*dz;
        // 2*(l-T)/l = 2 - 2*T*rsqrt(l2)
        float cf = (l2 > 1e-24f) ? bscale * (2.0f - 2.0f*TGT_BOND*__builtin_amdgcn_rsqf(l2))
                                 : 0.0f;
        float* g1 = g + (b*NATOM + n1)*3;
        float* g2 = g + (b*NATOM + n2)*3;
        atomicAdd(g2+0,  cf*dx); atomicAdd(g2+1,  cf*dy); atomicAdd(g2+2,  cf*dz);
        atomicAdd(g1+0, -cf*dx); atomicAdd(g1+1, -cf*dy); atomicAdd(g1+2, -cf*dz);
    } else if (tid < NBONDS + NANGS) {
        int t   = tid - NBONDS;
        int b   = t / NANG_PB;
        int rem = t % NANG_PB;
        int s   = rem / 3;
        int k   = rem % 3;                        // center atom = k+1
        int base = (b*NATOM + s*A_AT + k)*3;
        const float* p0 = x + base;
        const float* p1 = x + base + 3;
        const float* p2 = x + base + 6;
        float v1x=p0[0]-p1[0], v1y=p0[1]-p1[1], v1z=p0[2]-p1[2];
        float v2x=p2[0]-p1[0], v2y=p2[1]-p1[1], v2z=p2[2]-p1[2];
        float n1 = __builtin_amdgcn_sqrtf(v1x*v1x + v1y*v1y + v1z*v1z);
        float n2 = __builtin_amdgcn_sqrtf(v2x*v2x + v2y*v2y + v2z*v2z);
        float m1 = fmaxf(n1, 1e-8f), m2 = fmaxf(n2, 1e-8f);
        float dv = v1x*v2x + v1y*v2y + v1z*v2z;
        float cosr = dv * __builtin_amdgcn_rcpf(m1*m2);
        float cosc = fminf(fmaxf(cosr, -1.0f), 1.0f);
        float ang  = acosf(cosc);
        float sin2 = 1.0f - cosc*cosc;
        float gc = 0.0f;
        if (sin2 > 1e-12f && cosr > -1.0f && cosr < 1.0f)
            gc = -ascale * 2.0f * (ang - TGT_ANG) * __builtin_amdgcn_rsqf(sin2);
        float inv12 = __builtin_amdgcn_rcpf(m1*m2);
        float t1 = (n1 > 1e-8f) ? cosr * __builtin_amdgcn_rcpf(n1*m1) : 0.0f;
        float t2 = (n2 > 1e-8f) ? cosr * __builtin_amdgcn_rcpf(n2*m2) : 0.0f;
        float a0x = gc*(v2x*inv12 - v1x*t1), a0y = gc*(v2y*inv12 - v1y*t1), a0z = gc*(v2z*inv12 - v1z*t1);
        float a2x = gc*(v1x*inv12 - v2x*t2), a2y = gc*(v1y*inv12 - v2y*t2), a2z = gc*(v1z*inv12 - v2z*t2);
        float* g0 = g + base; float* g1 = g + base + 3; float* g2 = g + base + 6;
        atomicAdd(g0+0, a0x); atomicAdd(g0+1, a0y); atomicAdd(g0+2, a0z);
        atomicAdd(g2+0, a2x); atomicAdd(g2+1, a2y); atomicAdd(g2+2, a2z);
        atomicAdd(g1+0, -(a0x+a2x)); atomicAdd(g1+1, -(a0y+a2y)); atomicAdd(g1+2, -(a0z+a2z));
    }
}

// ---------------------------------------------------------------------------
// Adam update; consumes g and re-zeroes it for the next iteration
// ---------------------------------------------------------------------------
__global__ void gr_adam(float* __restrict__ x, float* __restrict__ m,
                        float* __restrict__ v, float* __restrict__ g,
                        float lr, float c1, float c2) {
    int i = blockIdx.x * blockDim.x + threadIdx.x;
    if (i < NCOORD) {
        float gi = g[i];
        float mi = 0.9f   * m[i] + 0.1f   * gi;
        float vi = 0.999f * v[i] + 0.001f * gi * gi;
        m[i] = mi; v[i] = vi;
        float mhat = mi / c1;
        float vhat = vi / c2;
        x[i] -= lr * mhat / (sqrtf(vhat) + 1e-8f);
        g[i] = 0.0f;
    }
}

// ---------------------------------------------------------------------------
extern "C" void kernel_launch(void* const* d_in, const int* in_sizes, int n_in,
                              void* d_out, int out_size, void* d_ws, size_t ws_size,
                              hipStream_t stream) {
    (void)in_sizes; (void)n_in; (void)out_size; (void)ws_size;
    const float* coords = (const float*)d_in[0];
    const float* restr  = (const float*)d_in[1];   // (2,256,256)
    // d_in[2] = mask: unused by the reference loss

    float* xw = (float*)d_ws;        // 7680
    float* gw = xw + NCOORD;         // 7680
    float* mw = gw + NCOORD;         // 7680
    float* vw = mw + NCOORD;         // 7680

    const float cscale = 2.0f * 20.0f / (2.0f * 1619250.0f); // 2*W_CLASH/(B*n_valid)
    const float rscale = 2.0f  / (2.0f * 32640.0f);          // W_REST /(B*n_pairs)
    const float bscale = 10.0f / (2.0f * 1275.0f);           // W_BOND /(B*n_bonds)
    const float ascale = 5.0f  / (2.0f * 768.0f);            // W_ANGLE/(B*S*3)

    gr_init<<<(NCOORD+127)/128, 128, 0, stream>>>(coords, xw, gw, mw, vw);

    float b1p = 1.0f, b2p = 1.0f;
    for (int t = 1; t <= 3; ++t) {
        b1p *= 0.9f; b2p *= 0.999f;
        gr_clash_grad<<<dim3(N_TILES/4, NB), 128, 0, stream>>>(xw, gw, cscale);
        gr_rest_grad <<<dim3(NR_TILES/4, NB), 128, 0, stream>>>(xw, restr, gw, rscale);
        gr_local_grad<<<(NBONDS + NANGS + 127)/128, 128, 0, stream>>>(xw, gw, bscale, ascale);
        gr_adam<<<(NCOORD+127)/128, 128, 0, stream>>>(xw, mw, vw, gw,
                                                      0.01f, 1.0f - b1p, 1.0f - b2p);
    }
    hipMemcpyAsync(d_out, xw, NCOORD*sizeof(float), hipMemcpyDeviceToDevice, stream);
}